// MultiHeadSelfAttention_44839458570543
// MI455X (gfx1250) — compile-verified
//
#include <hip/hip_runtime.h>
#include <math.h>

typedef float v2f __attribute__((ext_vector_type(2)));
typedef float v4f __attribute__((ext_vector_type(4)));
typedef float v8f __attribute__((ext_vector_type(8)));
typedef int   b128_t __attribute__((vector_size(16)));   // payload type for async b128

// Native fp32 tensor op: D(16x16) = A(16x4) * B(4x16) + C
#define WMMA_F32(a, b, c) \
  __builtin_amdgcn_wmma_f32_16x16x4_f32(false, (a), false, (b), (short)0, (c), false, false)

// ---- CDNA5 async global->LDS copy (ASYNCcnt-tracked, bypasses VGPRs) ------
#if __has_builtin(__builtin_amdgcn_global_load_async_to_lds_b128)
#define HAS_ASYNC_LDS 1
#else
#define HAS_ASYNC_LDS 0
#endif

__device__ __forceinline__ void g2l_b128(const float* g, float* l)
{
#if HAS_ASYNC_LDS
  __builtin_amdgcn_global_load_async_to_lds_b128(
      (__attribute__((address_space(1))) b128_t*)g,
      (__attribute__((address_space(3))) b128_t*)l, 0, 0);
#else
  *(v4f*)l = *(const v4f*)g;
#endif
}

#if __has_builtin(__builtin_amdgcn_s_wait_asynccnt)
#define WAIT_ASYNC(n) __builtin_amdgcn_s_wait_asynccnt(n)
#else
#define WAIT_ASYNC(n) asm volatile("s_wait_asynccnt " #n ::: "memory")
#endif

static constexpr int Bsz  = 2;
static constexpr int T    = 2048;
static constexpr int Cdim = 1024;
static constexpr int H    = 16;
static constexpr int Dh   = 64;           // head dim
static constexpr int Mtot = Bsz * T;      // 4096
static constexpr int F3   = 3 * Cdim;     // 3072
static constexpr int LDP  = 68;           // padded LDS row stride (floats)

// ---------------------------------------------------------------------------
// GEMM:  C[M,N] = A[M,K] * W[N,K]^T    (row-major A, row-major W)
// Block tile 128x64, 256 threads = 8 waves (4x2), wave tile 32x32 (2x2 wmma)
// Double-buffered LDS, async global->LDS prefetch (3 async ops / wave / tile)
// ---------------------------------------------------------------------------
__device__ __forceinline__ void stage_gemm_tile(const float* __restrict__ A,
                                                const float* __restrict__ W,
                                                float (*sA)[LDP], float (*sB)[LDP],
                                                int bm, int bn, int K, int kk, int tid)
{
#pragma unroll
  for (int it = 0; it < 2; ++it) {
    int idx = tid + it * 256;
    int r = idx >> 2, c = (idx & 3) * 4;
    g2l_b128(&A[(size_t)(bm + r) * K + kk + c], &sA[r][c]);
  }
  int r = tid >> 2, c = (tid & 3) * 4;
  g2l_b128(&W[(size_t)(bn + r) * K + kk + c], &sB[r][c]);
}

__global__ void __launch_bounds__(256)
gemm_xWt(const float* __restrict__ A, const float* __restrict__ W,
         float* __restrict__ Cout, int M, int N, int K)
{
  __shared__ float sA[2][128][LDP];
  __shared__ float sB[2][64][LDP];

  const int tid  = threadIdx.x;
  const int lane = tid & 31;
  const int wv   = tid >> 5;     // 0..7
  const int wm   = wv & 3;       // wave row (M)
  const int wn   = wv >> 2;      // wave col (N)
  const int lm   = lane & 15;
  const int lh   = lane >> 4;
  const int bm   = blockIdx.y * 128;
  const int bn   = blockIdx.x * 64;

  v8f acc[2][2];
#pragma unroll
  for (int i = 0; i < 2; ++i)
#pragma unroll
    for (int j = 0; j < 2; ++j)
#pragma unroll
      for (int e = 0; e < 8; ++e) acc[i][j][e] = 0.0f;

  const int ntiles = K / 16;
  stage_gemm_tile(A, W, sA[0], sB[0], bm, bn, K, 0, tid);

  for (int i = 0; i < ntiles; ++i) {
    if (i + 1 < ntiles) {
      // prefetch next K-slice into the other buffer (safe: end-barrier of
      // iteration i-1 guarantees no wave still reads buffer (i+1)&1)
      stage_gemm_tile(A, W, sA[(i + 1) & 1], sB[(i + 1) & 1],
                      bm, bn, K, (i + 1) * 16, tid);
      WAIT_ASYNC(3);            // tile i's 3 async ops have landed in LDS
    } else {
      WAIT_ASYNC(0);
    }
    __syncthreads();            // all waves' tile-i data visible

    const float (*cA)[LDP] = sA[i & 1];
    const float (*cB)[LDP] = sB[i & 1];
#pragma unroll
    for (int ks = 0; ks < 4; ++ks) {
      const int kf = ks * 4 + 2 * lh;
      v2f a0 = *(const v2f*)&cA[wm * 32 +  0 + lm][kf];
      v2f a1 = *(const v2f*)&cA[wm * 32 + 16 + lm][kf];
      v2f b0 = *(const v2f*)&cB[wn * 32 +  0 + lm][kf];
      v2f b1 = *(const v2f*)&cB[wn * 32 + 16 + lm][kf];
      acc[0][0] = WMMA_F32(a0, b0, acc[0][0]);
      acc[0][1] = WMMA_F32(a0, b1, acc[0][1]);
      acc[1][0] = WMMA_F32(a1, b0, acc[1][0]);
      acc[1][1] = WMMA_F32(a1, b1, acc[1][1]);
    }
    __syncthreads();            // everyone done reading buffer i&1
  }

  // epilogue: C-layout element (M = g + 8*(lane/16), N = lane%16)
#pragma unroll
  for (int i = 0; i < 2; ++i)
#pragma unroll
    for (int j = 0; j < 2; ++j)
#pragma unroll
      for (int g = 0; g < 8; ++g) {
        int row = bm + wm * 32 + i * 16 + g + 8 * lh;
        int col = bn + wn * 32 + j * 16 + lm;
        Cout[(size_t)row * N + col] = acc[i][j][g];
      }
}

// ---------------------------------------------------------------------------
// RoPE on q and k halves of qkv (in place). Each thread owns a (d, d+32) pair
// of one head for both q and k, so rotate-half has no read/write race.
// ---------------------------------------------------------------------------
__global__ void __launch_bounds__(256)
rope_qk(float* __restrict__ qkv, const float* __restrict__ cosp,
        const float* __restrict__ sinp)
{
  int e = blockIdx.x * blockDim.x + threadIdx.x;     // [0, Mtot*512)
  if (e >= Mtot * (Cdim / 2)) return;
  int m = e / (Cdim / 2);
  int p = e % (Cdim / 2);
  int h = p / (Dh / 2);
  int d = p % (Dh / 2);
  int t = m % T;

  float c0 = cosp[t * Dh + d],      s0 = sinp[t * Dh + d];
  float c1 = cosp[t * Dh + d + 32], s1 = sinp[t * Dh + d + 32];

  size_t base = (size_t)m * F3 + h * Dh + d;
  float q0 = qkv[base], q1 = qkv[base + 32];
  qkv[base]      = q0 * c0 - q1 * s0;
  qkv[base + 32] = q1 * c1 + q0 * s1;

  float k0 = qkv[base + Cdim], k1 = qkv[base + Cdim + 32];
  qkv[base + Cdim]      = k0 * c0 - k1 * s0;
  qkv[base + Cdim + 32] = k1 * c1 + k0 * s1;
}

// ---------------------------------------------------------------------------
// Causal flash attention. Block = (query tile of 64, b*H+h), 128 threads =
// 4 waves; each wave owns 16 query rows. K/V tiles double-buffered with
// async global->LDS prefetch (16 async ops / wave / key tile).
// ---------------------------------------------------------------------------
__device__ __forceinline__ void stage_kv(const float* __restrict__ Kg,
                                         const float* __restrict__ Vg,
                                         float (*sK)[LDP], float (*sV)[LDP],
                                         int kBase, int tid)
{
#pragma unroll
  for (int it = 0; it < 8; ++it) {
    int idx = tid + it * 128;
    int r = idx >> 4, c = (idx & 15) * 4;
    g2l_b128(&Kg[(size_t)(kBase + r) * F3 + c], &sK[r][c]);
    g2l_b128(&Vg[(size_t)(kBase + r) * F3 + c], &sV[r][c]);
  }
}

__global__ void __launch_bounds__(128)
flash_attn(const float* __restrict__ qkv, float* __restrict__ Y)
{
  __shared__ float sQ[64][LDP];
  __shared__ float sK[2][64][LDP];
  __shared__ float sV[2][64][LDP];
  __shared__ float sP[4][16][LDP];   // wave-private P staging (C-layout -> A-layout)

  const int tid  = threadIdx.x;
  const int lane = tid & 31;
  const int wv   = tid >> 5;        // 0..3
  const int lm   = lane & 15;
  const int lh   = lane >> 4;
  const int tq   = blockIdx.x;      // query tile index (0..31)
  const int bh   = blockIdx.y;      // b*H + h
  const int b    = bh / H, h = bh % H;
  const int qBase = tq * 64;
  const float scale = 0.125f;       // 1/sqrt(64)

  const float* Qg = qkv + (size_t)(b * T) * F3 + h * Dh;
  const float* Kg = Qg + Cdim;
  const float* Vg = Qg + 2 * Cdim;

  // async-stage Q tile and the first K/V tile
#pragma unroll
  for (int it = 0; it < 8; ++it) {
    int idx = tid + it * 128;                 // 0..1023
    int r = idx >> 4, c = (idx & 15) * 4;
    g2l_b128(&Qg[(size_t)(qBase + r) * F3 + c], &sQ[r][c]);
  }
  stage_kv(Kg, Vg, sK[0], sV[0], 0, tid);

  v8f o[4];
  float rmax[8], rsum[8];
#pragma unroll
  for (int t_ = 0; t_ < 4; ++t_)
#pragma unroll
    for (int e = 0; e < 8; ++e) o[t_][e] = 0.0f;
#pragma unroll
  for (int g = 0; g < 8; ++g) { rmax[g] = -1e30f; rsum[g] = 0.0f; }

  for (int kt = 0; kt <= tq; ++kt) {
    const int kBase = kt * 64;
    const int cur = kt & 1;
    if (kt + 1 <= tq) {
      stage_kv(Kg, Vg, sK[cur ^ 1], sV[cur ^ 1], (kt + 1) * 64, tid);
      WAIT_ASYNC(16);           // current tile (and Q on kt==0) has landed
    } else {
      WAIT_ASYNC(0);
    }
    __syncthreads();

    // S(16x64) = Qw(16x64) * K^T(64x64) : 4 N-tiles, 16 k-steps of 4
    v8f s[4];
#pragma unroll
    for (int t_ = 0; t_ < 4; ++t_)
#pragma unroll
      for (int e = 0; e < 8; ++e) s[t_][e] = 0.0f;

#pragma unroll
    for (int dsv = 0; dsv < 16; ++dsv) {
      const int kf = dsv * 4 + 2 * lh;
      v2f a = *(const v2f*)&sQ[wv * 16 + lm][kf];
#pragma unroll
      for (int t_ = 0; t_ < 4; ++t_) {
        v2f bf = *(const v2f*)&sK[cur][t_ * 16 + lm][kf];
        s[t_] = WMMA_F32(a, bf, s[t_]);
      }
    }

    // softmax scale (folded here instead of pre-scaling Q)
#pragma unroll
    for (int t_ = 0; t_ < 4; ++t_) s[t_] *= scale;

    // causal mask on the diagonal tile
    if (kt == tq) {
#pragma unroll
      for (int t_ = 0; t_ < 4; ++t_) {
        int key = kBase + t_ * 16 + lm;
#pragma unroll
        for (int g = 0; g < 8; ++g) {
          int qrow = qBase + wv * 16 + g + 8 * lh;
          if (key > qrow) s[t_][g] = -1e30f;
        }
      }
    }

    // online softmax per row g (row lives in 16 lanes of one half-wave)
#pragma unroll
    for (int g = 0; g < 8; ++g) {
      float tm = fmaxf(fmaxf(s[0][g], s[1][g]), fmaxf(s[2][g], s[3][g]));
      tm = fmaxf(tm, __shfl_xor(tm, 1));
      tm = fmaxf(tm, __shfl_xor(tm, 2));
      tm = fmaxf(tm, __shfl_xor(tm, 4));
      tm = fmaxf(tm, __shfl_xor(tm, 8));
      float mnew = fmaxf(rmax[g], tm);
      float corr = __expf(rmax[g] - mnew);
      rmax[g] = mnew;
      float ps = 0.0f;
#pragma unroll
      for (int t_ = 0; t_ < 4; ++t_) {
        float pv = __expf(s[t_][g] - mnew);
        s[t_][g] = pv;
        ps += pv;
      }
      ps += __shfl_xor(ps, 1);
      ps += __shfl_xor(ps, 2);
      ps += __shfl_xor(ps, 4);
      ps += __shfl_xor(ps, 8);
      rsum[g] = rsum[g] * corr + ps;
#pragma unroll
      for (int t_ = 0; t_ < 4; ++t_) o[t_][g] *= corr;
    }

    // C-layout P -> LDS (wave-private, no block barrier needed)
#pragma unroll
    for (int t_ = 0; t_ < 4; ++t_)
#pragma unroll
      for (int g = 0; g < 8; ++g)
        sP[wv][g + 8 * lh][t_ * 16 + lm] = s[t_][g];

    // O(16x64) += P(16x64) * V(64x64)
#pragma unroll
    for (int kkv = 0; kkv < 16; ++kkv) {
      const int kf = kkv * 4 + 2 * lh;
      v2f a = *(const v2f*)&sP[wv][lm][kf];
#pragma unroll
      for (int t2 = 0; t2 < 4; ++t2) {
        v2f bf;
        bf[0] = sV[cur][kf + 0][t2 * 16 + lm];
        bf[1] = sV[cur][kf + 1][t2 * 16 + lm];
        o[t2] = WMMA_F32(a, bf, o[t2]);
      }
    }
    __syncthreads();            // everyone done reading sK/sV[cur] before refill
  }

  // normalize and write y[m, h*64 + d]
#pragma unroll
  for (int g = 0; g < 8; ++g) {
    float inv = 1.0f / rsum[g];
    int row = b * T + qBase + wv * 16 + g + 8 * lh;
#pragma unroll
    for (int t_ = 0; t_ < 4; ++t_)
      Y[(size_t)row * Cdim + h * Dh + t_ * 16 + lm] = o[t_][g] * inv;
  }
}

// ---------------------------------------------------------------------------
extern "C" void kernel_launch(void* const* d_in, const int* in_sizes, int n_in,
                              void* d_out, int out_size, void* d_ws, size_t ws_size,
                              hipStream_t stream)
{
  const float* x    = (const float*)d_in[0];
  const float* cosp = (const float*)d_in[1];
  const float* sinp = (const float*)d_in[2];
  const float* Wqkv = (const float*)d_in[3];
  const float* Wout = (const float*)d_in[4];
  float* out = (float*)d_out;

  float* qkv = (float*)d_ws;                       // Mtot * 3072 floats (50.3 MB)
  float* y   = qkv + (size_t)Mtot * F3;            // Mtot * 1024 floats (16.8 MB)

  // 1) qkv = x @ Wqkv^T
  gemm_xWt<<<dim3(F3 / 64, Mtot / 128), 256, 0, stream>>>(x, Wqkv, qkv, Mtot, F3, Cdim);
  // 2) RoPE on q,k
  rope_qk<<<dim3((Mtot * (Cdim / 2)) / 256), 256, 0, stream>>>(qkv, cosp, sinp);
  // 3) causal flash attention -> y
  flash_attn<<<dim3(T / 64, Bsz * H), 128, 0, stream>>>(qkv, y);
  // 4) out = y @ Wout^T
  gemm_xWt<<<dim3(Cdim / 64, Mtot / 128), 256, 0, stream>>>(y, Wout, out, Mtot, Cdim, Cdim);
}